// KGEModel_12833362280951
// MI455X (gfx1250) — compile-verified
//
#include <hip/hip_runtime.h>

#define NENTITY   200000
#define NRELATION 500
#define DIM       512
#define BATCH     256
#define NEG       1024

#define WAVES_PER_BLOCK 8
#define NPW             16   // negatives per wave
#define NBUF            3    // LDS ring buffers per wave (2-row lookahead)
// block = 256 threads (8 wave32), grid = (BATCH, NEG / (WAVES_PER_BLOCK*NPW)) = (256, 8)

typedef int v4i __attribute__((ext_vector_type(4)));

#if defined(__AMDGCN__) && __has_builtin(__builtin_amdgcn_global_load_async_to_lds_b128)
#define USE_ASYNC 1
#else
#define USE_ASYNC 0
#endif

#if __has_builtin(__builtin_amdgcn_s_wait_asynccnt)
#define WAIT_ASYNC(n) __builtin_amdgcn_s_wait_asynccnt(n)
#else
#define WAIT_ASYNC(n) asm volatile("s_wait_asynccnt %0" ::"i"(n) : "memory")
#endif

// generic -> concrete address-space casts (via integer; low 32 bits of a
// generic LDS pointer are the LDS byte offset on amdgcn).
// The async builtin wants non-const pointers to v4i in AS1 (global) / AS3 (LDS).
#define TO_AS1(p) ((__attribute__((address_space(1))) v4i*)(unsigned long long)(p))
#define TO_AS3(p) ((__attribute__((address_space(3))) v4i*)(unsigned int)(unsigned long long)(p))

__global__ __launch_bounds__(256) void transe_headbatch_kernel(
    const int*   __restrict__ tail_part,   // [B,3]
    const int*   __restrict__ head_part,   // [B,NEG]
    const float* __restrict__ ent,         // [NENTITY, DIM]
    const float* __restrict__ rel,         // [NRELATION, DIM]
    const float* __restrict__ gamma,       // [1]
    float*       __restrict__ out)         // [B, NEG]
{
    const int b    = blockIdx.x;
    const int tid  = threadIdx.x;
    const int wave = tid >> 5;
    const int lane = tid & 31;

    __shared__ float c[DIM];
#if USE_ASYNC
    __shared__ float hbuf[WAVES_PER_BLOCK][NBUF][DIM];
#endif

    // c[d] = r[d] - t[d]; DIM=512, 256 threads -> exactly two strided writes
    const int rid  = tail_part[b * 3 + 1];
    const int tide = tail_part[b * 3 + 2];
    {
        const float* rrow = rel + rid * DIM;
        const float* trow = ent + (long long)tide * DIM;
        c[tid]       = rrow[tid]       - trow[tid];
        c[tid + 256] = rrow[tid + 256] - trow[tid + 256];
    }
    __syncthreads();

    // each lane caches its 16-float slice of c: floats [lane*4 + j*128, +4)
    float4 creg[4];
#pragma unroll
    for (int j = 0; j < 4; ++j)
        creg[j] = *(const float4*)(c + lane * 4 + j * 128);

    const float g = gamma[0];
    const int nbase = blockIdx.y * (WAVES_PER_BLOCK * NPW) + wave * NPW;
    const int* __restrict__ hp = head_part + b * NEG + nbase;  // uniform per wave -> SGPRs
    float* __restrict__ outp = out + b * NEG + nbase;

#if USE_ASYNC
    float* const bufs[NBUF] = {
        &hbuf[wave][0][0], &hbuf[wave][1][0], &hbuf[wave][2][0]
    };

    // async prefetch of one 2KB head row: 4 x b128 per lane, 512B contiguous
    // per instruction across the wave (perfectly coalesced global reads)
    auto prefetch = [&](int i, float* dst) {
        const unsigned long long hid = (unsigned int)hp[i];
        const char* grow = (const char*)(ent + hid * (unsigned long long)DIM) + lane * 16;
        char* lrow = (char*)dst + lane * 16;
#pragma unroll
        for (int j = 0; j < 4; ++j)
            __builtin_amdgcn_global_load_async_to_lds_b128(
                TO_AS1(grow + j * 512), TO_AS3(lrow + j * 512), 0, 0);
    };

    prefetch(0, bufs[0]);
    prefetch(1, bufs[1]);
#pragma unroll
    for (int i = 0; i < NPW; ++i) {
        if (i + 2 < NPW) prefetch(i + 2, bufs[(i + 2) % NBUF]);
        // async loads complete in order; 4 ops per row group
        if (i + 2 < NPW)      WAIT_ASYNC(8);   // rows i+1, i+2 may be in flight
        else if (i + 1 < NPW) WAIT_ASYNC(4);   // row i+1 may be in flight
        else                  WAIT_ASYNC(0);
        asm volatile("" ::: "memory");         // no hoisting of LDS reads above the wait

        const float4* hb = (const float4*)bufs[i % NBUF] + lane;
        float s = 0.f;
#pragma unroll
        for (int j = 0; j < 4; ++j) {
            float4 hv = hb[j * 32];            // ds_load_b128, matches creg slice
            s += fabsf(hv.x + creg[j].x) + fabsf(hv.y + creg[j].y) +
                 fabsf(hv.z + creg[j].z) + fabsf(hv.w + creg[j].w);
        }
#pragma unroll
        for (int m = 16; m >= 1; m >>= 1) s += __shfl_xor(s, m, 32);
        if (lane == 0) outp[i] = g - s;
    }
#else
    // fallback: direct coalesced global b128 loads
    for (int i = 0; i < NPW; ++i) {
        const unsigned long long hid = (unsigned int)hp[i];
        const float4* hrow = (const float4*)(ent + hid * (unsigned long long)DIM);
        float s = 0.f;
#pragma unroll
        for (int j = 0; j < 4; ++j) {
            float4 hv = hrow[lane + j * 32];
            s += fabsf(hv.x + creg[j].x) + fabsf(hv.y + creg[j].y) +
                 fabsf(hv.z + creg[j].z) + fabsf(hv.w + creg[j].w);
        }
#pragma unroll
        for (int m = 16; m >= 1; m >>= 1) s += __shfl_xor(s, m, 32);
        if (lane == 0) outp[i] = g - s;
    }
#endif
}

extern "C" void kernel_launch(void* const* d_in, const int* in_sizes, int n_in,
                              void* d_out, int out_size, void* d_ws, size_t ws_size,
                              hipStream_t stream) {
    (void)in_sizes; (void)n_in; (void)out_size; (void)d_ws; (void)ws_size;
    const int*   tail_part = (const int*)d_in[0];
    const int*   head_part = (const int*)d_in[1];
    const float* ent       = (const float*)d_in[2];
    const float* rel       = (const float*)d_in[3];
    const float* gamma     = (const float*)d_in[4];
    float*       out       = (float*)d_out;

    dim3 grid(BATCH, NEG / (WAVES_PER_BLOCK * NPW));  // (256, 8)
    dim3 block(256);
    transe_headbatch_kernel<<<grid, block, 0, stream>>>(
        tail_part, head_part, ent, rel, gamma, out);
}